// gcomreopool_62792421868033
// MI455X (gfx1250) — compile-verified
//
#include <hip/hip_runtime.h>
#include <stdint.h>

#define NB   8
#define NROW 4096
#define NF   128
#define GS   1024
#define RPB  2          // A-rows gathered per block in kernel 2

// ---- CDNA5 async global->LDS path (probe via __has_builtin; fallback = float4) ----
#if defined(__has_builtin)
#  if __has_builtin(__builtin_amdgcn_global_load_async_to_lds_b128)
#    define USE_ASYNC_LDS 1
#  endif
#  if __has_builtin(__builtin_nontemporal_store)
#    define HAVE_NT 1
#  endif
#endif
#ifndef USE_ASYNC_LDS
#  define USE_ASYNC_LDS 0
#endif
#ifndef HAVE_NT
#  define HAVE_NT 0
#endif

// Builtin signature (leaked by clang diagnostic): param0 is
// 'int __attribute__((vector_size(16))) __device__ *'  == int4 addrspace(1)*.
typedef int   v4i __attribute__((ext_vector_type(4)));
typedef float v4f __attribute__((ext_vector_type(4)));
typedef __attribute__((address_space(1))) v4i glb_v4i;
typedef __attribute__((address_space(3))) v4i lds_v4i;

__device__ __forceinline__ void nt_store_f32(float* p, float v) {
#if HAVE_NT
  __builtin_nontemporal_store(v, p);
#else
  *p = v;
#endif
}
__device__ __forceinline__ void nt_store_v4f(v4f* p, v4f v) {
#if HAVE_NT
  __builtin_nontemporal_store(v, p);
#else
  *p = v;
#endif
}
__device__ __forceinline__ v4f nt_load_v4f(const v4f* p) {
#if HAVE_NT
  return __builtin_nontemporal_load(p);
#else
  return *p;
#endif
}

__device__ __forceinline__ uint32_t orderable_f32(float f) {
  uint32_t u = __float_as_uint(f);
  // monotone float -> uint map (ascending)
  return (u & 0x80000000u) ? ~u : (u | 0x80000000u);
}

// ---------------------------------------------------------------------------
// Kernel 1: per-batch top-1024 via full bitonic sort of 4096 keys in LDS.
// Key = (~orderable(val) << 32) | index  -> ascending sort == top_k order
// (value descending, ties broken by ascending index, matching jax.lax.top_k).
// Also performs the xg gather: one wave32 per row (32 lanes x float4 = 128 f).
// ---------------------------------------------------------------------------
__global__ __launch_bounds__(1024) void topk_sort_kernel(
    const float* __restrict__ x,       // (NB, NROW, NF)
    int*         __restrict__ idx_out, // (NB, GS)
    float*       __restrict__ xg)      // (NB, GS, NF)
{
  __shared__ uint64_t key[NROW];   // 32 KiB
  const int b = blockIdx.x;
  const int t = threadIdx.x;
  const float* xb = x + (size_t)b * NROW * NF;

  for (int i = t; i < NROW; i += 1024) {
    float v = xb[(size_t)i * NF + (NF - 1)];
    uint32_t ou = orderable_f32(v);
    key[i] = ((uint64_t)(~ou) << 32) | (uint32_t)i;
  }
  __syncthreads();

  // Bitonic sort, ascending.
  for (int k = 2; k <= NROW; k <<= 1) {
    for (int j = k >> 1; j > 0; j >>= 1) {
      for (int i = t; i < NROW; i += 1024) {
        int ixj = i ^ j;
        if (ixj > i) {
          uint64_t a = key[i], c = key[ixj];
          bool up = ((i & k) == 0);
          if (up ? (a > c) : (a < c)) { key[i] = c; key[ixj] = a; }
        }
      }
      __syncthreads();
    }
  }

  // Emit sorted indices (front 1024 == top_k order).
  int my_idx = (int)(uint32_t)(key[t] & 0xFFFFFFFFull);
  if (t < GS) idx_out[(size_t)b * GS + t] = my_idx;

  // xg gather: wave w handles rows w, w+32, ... ; lane = one float4 of the row.
  // Rows are read once and outputs written once -> non-temporal on both sides.
  const int lane = t & 31;
  const int wave = t >> 5;            // 0..31
  for (int r = wave; r < GS; r += 32) {
    int src = (int)(uint32_t)(key[r] & 0xFFFFFFFFull);
    const v4f* sp = (const v4f*)(xb + (size_t)src * NF);
    v4f v = nt_load_v4f(sp + lane);
    nt_store_v4f((v4f*)(xg + ((size_t)b * GS + r) * NF) + lane, v);
  }
}

// ---------------------------------------------------------------------------
// Kernel 2: At2[b,i,j] = A[b, idx[i], idx[j]].
// Stage RPB full 16 KiB A-rows into LDS (async DMA when available), then
// resolve the random column gather from LDS and store coalesced + non-temporal.
// ---------------------------------------------------------------------------
__global__ __launch_bounds__(256) void row_gather_kernel(
    const float* __restrict__ A,    // (NB, NROW, NROW)
    const int*   __restrict__ idx,  // (NB, GS)
    float*       __restrict__ out)  // (NB, GS, GS)
{
  __shared__ float rows[RPB][NROW];  // RPB * 16 KiB
  __shared__ int   cols[GS];         // 4 KiB
  const int b  = blockIdx.y;
  const int i0 = blockIdx.x * RPB;
  const int t  = threadIdx.x;
  const int* idxb = idx + (size_t)b * GS;

  for (int j = t; j < GS; j += 256) cols[j] = idxb[j];

  for (int rr = 0; rr < RPB; ++rr) {
    const int r = idxb[i0 + rr];  // uniform -> scalar load
    const float* rowp = A + ((size_t)b * NROW + (size_t)r) * NROW;
#if USE_ASYNC_LDS
    for (int c = t; c < NROW / 4; c += 256) {
      __builtin_amdgcn_global_load_async_to_lds_b128(
          (glb_v4i*)(rowp + 4 * c),
          (lds_v4i*)(&rows[rr][4 * c]),
          0, 0);
    }
#else
    for (int c = t; c < NROW / 4; c += 256) {
      ((v4f*)rows[rr])[c] = ((const v4f*)rowp)[c];
    }
#endif
  }
#if USE_ASYNC_LDS
#  if defined(__has_builtin) && __has_builtin(__builtin_amdgcn_s_wait_asynccnt)
  __builtin_amdgcn_s_wait_asynccnt(0);
#  else
  asm volatile("s_wait_asynccnt 0" ::: "memory");
#  endif
#endif
  __syncthreads();

  for (int rr = 0; rr < RPB; ++rr) {
    float* op = out + ((size_t)b * GS + (size_t)(i0 + rr)) * GS;
    for (int j = t; j < GS; j += 256) {
      nt_store_f32(op + j, rows[rr][cols[j]]);
    }
  }
}

// ---------------------------------------------------------------------------
extern "C" void kernel_launch(void* const* d_in, const int* in_sizes, int n_in,
                              void* d_out, int out_size, void* d_ws, size_t ws_size,
                              hipStream_t stream) {
  (void)in_sizes; (void)n_in; (void)out_size; (void)ws_size;
  const float* A = (const float*)d_in[0];   // (8, 4096, 4096) f32
  const float* x = (const float*)d_in[1];   // (8, 4096, 128)  f32

  float* At2 = (float*)d_out;                                  // (8,1024,1024)
  float* xg  = (float*)d_out + (size_t)NB * GS * GS;           // (8,1024,128)
  int*   idx = (int*)d_ws;                                     // (8,1024) scratch

  topk_sort_kernel<<<dim3(NB), dim3(1024), 0, stream>>>(x, idx, xg);
  row_gather_kernel<<<dim3(GS / RPB, NB), dim3(256), 0, stream>>>(A, idx, At2);
}